// PiecewiseLinearShapeNN2D_19095424598302
// MI455X (gfx1250) — compile-verified
//
#include <hip/hip_runtime.h>
#include <stdint.h>

// Memory-bound FEM gather/eval kernel for MI455X (gfx1250, wave32).
// WMMA is deliberately NOT used: per-point 3x2 operands admit no shared
// matmul structure, and at ~0.4 flop/byte the kernel is HBM/L2 bound.
// CDNA5 features used:
//  - GLOBAL_LOAD_ASYNC_TO_LDS_B64 (ASYNCcnt) to stage the x_eval stream into
//    LDS while the 24 random gathers are in flight; consumed after
//    s_wait_asynccnt 0. Lane-private slots -> no workgroup barrier needed.
//  - non-temporal TH hints on once-touched streams (protects L2-resident
//    gather tables: conn/coords/u, 40 MB, hit 14M times from 192 MB L2).
//  - global_prefetch_b8 on the sequential streams.

typedef __attribute__((ext_vector_type(2))) float f32x2;
typedef __attribute__((ext_vector_type(4))) float f32x4;

#define THREADS 256
#define PT 4
#define TILE (THREADS * PT)
#define PF_DIST (4 * TILE)   // prefetch ~4 tiles (4096 points) ahead

__device__ __forceinline__ uint32_t lds_addr32(const void* p) {
    // LDS aperture mapping: logical LDS byte address = flat_addr[31:0]
    return (uint32_t)(uintptr_t)p;
}

__device__ __forceinline__ void async_load_b64_to_lds(uint32_t lds_byte_addr,
                                                      uint32_t glb_byte_off,
                                                      const void* glb_base) {
    asm volatile("global_load_async_to_lds_b64 %0, %1, %2 th:TH_LOAD_NT"
                 :
                 : "v"(lds_byte_addr), "v"(glb_byte_off), "s"(glb_base)
                 : "memory");
}

__device__ __forceinline__ void wait_asynccnt0() {
#if __has_builtin(__builtin_amdgcn_s_wait_asynccnt)
    __builtin_amdgcn_s_wait_asynccnt(0);
#else
    asm volatile("s_wait_asynccnt 0x0" ::: "memory");
#endif
}

__global__ __launch_bounds__(THREADS) void fem_eval_kernel(
    const f32x2* __restrict__ x_eval,
    const f32x2* __restrict__ coords,
    const f32x2* __restrict__ uu,
    const int*   __restrict__ elem_id,
    const int*   __restrict__ conn,
    f32x2*       __restrict__ out_uh,
    float*       __restrict__ out_det,
    f32x4*       __restrict__ out_grad,
    int M)
{
    __shared__ f32x2 s_x[TILE];   // 8 KB staging for the x_eval stream

    const int t    = threadIdx.x;
    const int tile = blockIdx.x * TILE;

    // gfx1250 global_prefetch_b8: warm GL2 ahead of the sequential streams.
    {
        const int pf = tile + PF_DIST + t;
        if (pf < M) {
            __builtin_prefetch(&elem_id[pf], 0, 0);
            __builtin_prefetch(&x_eval[pf], 0, 0);
        }
    }

    int  idx[PT];
    bool ok[PT];
    int  e[PT];

    const uint32_t lds0 = lds_addr32(&s_x[0]);

    // Phase 1: kick off async LDS staging of x_eval (consumed last, so it
    // overlaps the entire gather chain), and load elem_id (needed now).
#pragma unroll
    for (int k = 0; k < PT; ++k) {
        idx[k] = tile + k * THREADS + t;
        ok[k]  = idx[k] < M;
        const int ii = ok[k] ? idx[k] : 0;
        async_load_b64_to_lds(lds0 + (uint32_t)(k * THREADS + t) * 8u,
                              (uint32_t)ii * 8u, x_eval);
        e[k] = __builtin_nontemporal_load(&elem_id[ii]);
    }

    // Phase 2: connectivity gathers (regular temporal — L2 resident, reused).
    int n0[PT], n1[PT], n2[PT];
#pragma unroll
    for (int k = 0; k < PT; ++k) {
        const int b = 3 * e[k];
        n0[k] = conn[b + 0];
        n1[k] = conn[b + 1];
        n2[k] = conn[b + 2];
    }

    // Phase 3: node-data gathers, all in flight together (24 b64 loads/thread).
    f32x2 v0[PT], v1[PT], v2[PT], u0[PT], u1[PT], u2[PT];
#pragma unroll
    for (int k = 0; k < PT; ++k) {
        v0[k] = coords[n0[k]];
        v1[k] = coords[n1[k]];
        v2[k] = coords[n2[k]];
        u0[k] = uu[n0[k]];
        u1[k] = uu[n1[k]];
        u2[k] = uu[n2[k]];
    }

    // x_eval staging has had the whole gather phase to land.
    wait_asynccnt0();

    // Phase 4: ~25 VALU ops per point, then non-temporal stores.
#pragma unroll
    for (int k = 0; k < PT; ++k) {
        if (!ok[k]) continue;

        const f32x2 x  = s_x[k * THREADS + t];   // ds_load_b64, lane-private
        const float xi = x.x, eta = x.y;
        const float N2 = 1.0f - xi - eta;
        const f32x2 uh = xi * u0[k] + eta * u1[k] + N2 * u2[k];

        const float a = v1[k].x - v0[k].x;   // e1.x
        const float c = v1[k].y - v0[k].y;   // e1.y
        const float b = v2[k].x - v0[k].x;   // e2.x
        const float d = v2[k].y - v0[k].y;   // e2.y
        const float det = a * d - b * c;
        const float inv = 1.0f / det;

        // grad_u[j][i] = Jinv[i][0]*du1[j] + Jinv[i][1]*du2[j],
        // Jinv = [[d,-b],[-c,a]] * inv
        const f32x2 du1 = u1[k] - u0[k];
        const f32x2 du2 = u2[k] - u0[k];
        f32x4 g;
        g.x = ( d * du1.x - b * du2.x) * inv;   // grad_u[0][0]
        g.y = (-c * du1.x + a * du2.x) * inv;   // grad_u[0][1]
        g.z = ( d * du1.y - b * du2.y) * inv;   // grad_u[1][0]
        g.w = (-c * du1.y + a * du2.y) * inv;   // grad_u[1][1]

        __builtin_nontemporal_store(uh,  &out_uh[idx[k]]);
        __builtin_nontemporal_store(det, &out_det[idx[k]]);
        __builtin_nontemporal_store(g,   &out_grad[idx[k]]);
    }
}

extern "C" void kernel_launch(void* const* d_in, const int* in_sizes, int n_in,
                              void* d_out, int out_size, void* d_ws, size_t ws_size,
                              hipStream_t stream)
{
    const f32x2* x_eval  = (const f32x2*)d_in[0];   // (M,2) f32
    const f32x2* coords  = (const f32x2*)d_in[1];   // (N,2) f32
    const f32x2* u       = (const f32x2*)d_in[2];   // (N,2) f32
    const int*   elem_id = (const int*)  d_in[3];   // (M,)  i32
    const int*   conn    = (const int*)  d_in[4];   // (E,3) i32

    const int M = in_sizes[3];

    float* out      = (float*)d_out;
    f32x2* out_uh   = (f32x2*)out;                         // M x 2
    float* out_det  = out + (size_t)2 * M;                 // M
    f32x4* out_grad = (f32x4*)(out + (size_t)3 * M);       // M x 2 x 2

    const int grid = (M + TILE - 1) / TILE;
    fem_eval_kernel<<<grid, THREADS, 0, stream>>>(
        x_eval, coords, u, elem_id, conn, out_uh, out_det, out_grad, M);
}